// GCN_3453153706769
// MI455X (gfx1250) — compile-verified
//
#include <hip/hip_runtime.h>
#include <math.h>

typedef float v2f __attribute__((ext_vector_type(2)));
typedef float v8f __attribute__((ext_vector_type(8)));

#define N_NODES   100000
#define N_FEAT    512
#define HIDDEN    64
#define N_CLASSES 16

static __device__ __forceinline__ void atomic_add_f32(float* p, float v) {
    __hip_atomic_fetch_add(p, v, __ATOMIC_RELAXED, __HIP_MEMORY_SCOPE_AGENT);
}

// ---------------------------------------------------------------------------
// init: bufB (h1 accumulator) = 0, d_out (h2 accumulator) = 0, deg = 1 (self-loop)
// ---------------------------------------------------------------------------
__global__ __launch_bounds__(256) void init_kernel(float* bufB, float* out, float* deg) {
    int t = blockIdx.x * 256 + threadIdx.x;
    if (t < N_NODES * HIDDEN)    bufB[t] = 0.0f;
    if (t < N_NODES * N_CLASSES) out[t]  = 0.0f;
    if (t < N_NODES)             deg[t]  = 1.0f;
}

// ---------------------------------------------------------------------------
// degree: deg[dst[e]] += 1
// ---------------------------------------------------------------------------
__global__ __launch_bounds__(256) void degree_kernel(const int* __restrict__ dst,
                                                     float* __restrict__ deg, int nE) {
    int e = blockIdx.x * 256 + threadIdx.x;
    if (e < nE) atomic_add_f32(&deg[dst[e]], 1.0f);
}

// dinv = rsqrt(deg)  (deg >= 1 always due to self-loop)
__global__ __launch_bounds__(256) void rsqrt_kernel(float* dinv) {
    int t = blockIdx.x * 256 + threadIdx.x;
    if (t < N_NODES) dinv[t] = rsqrtf(dinv[t]);
}

// ---------------------------------------------------------------------------
// GEMM1: C[100000x64] = A[100000x512] * B[512x64]   (fp32 WMMA 16x16x4)
// one wave computes a 16-row stripe across all 64 output columns (4 N-tiles)
// ---------------------------------------------------------------------------
__global__ __launch_bounds__(256) void gemm1_kernel(const float* __restrict__ A,
                                                    const float* __restrict__ B,
                                                    float* __restrict__ C, int nTiles) {
    int wave = threadIdx.x >> 5;
    int lane = threadIdx.x & 31;
    int tile = blockIdx.x * 8 + wave;
    if (tile >= nTiles) return;                 // whole-wave exit: EXEC all-1 for WMMA
    int nr = lane & 15;                         // row (A) / col (B,C) within tile
    int hi = lane >> 4;                         // K half selector

    const float* arow = A + (size_t)(tile * 16 + nr) * N_FEAT + 2 * hi;
    v8f acc0 = {}, acc1 = {}, acc2 = {}, acc3 = {};

    for (int kk = 0; kk < N_FEAT; kk += 4) {
        v2f a = *(const v2f*)(arow + kk);       // A[m][kk+2hi], A[m][kk+2hi+1]
        const float* b0p = B + (size_t)(kk + 2 * hi) * HIDDEN + nr;
        const float* b1p = b0p + HIDDEN;
        v2f b0 = { b0p[0],  b1p[0]  };
        v2f b1 = { b0p[16], b1p[16] };
        v2f b2 = { b0p[32], b1p[32] };
        v2f b3 = { b0p[48], b1p[48] };
        acc0 = __builtin_amdgcn_wmma_f32_16x16x4_f32(false, a, false, b0, (short)0, acc0, false, false);
        acc1 = __builtin_amdgcn_wmma_f32_16x16x4_f32(false, a, false, b1, (short)0, acc1, false, false);
        acc2 = __builtin_amdgcn_wmma_f32_16x16x4_f32(false, a, false, b2, (short)0, acc2, false, false);
        acc3 = __builtin_amdgcn_wmma_f32_16x16x4_f32(false, a, false, b3, (short)0, acc3, false, false);
    }
    int mbase = tile * 16 + hi * 8;             // C layout: VGPR r -> row r (lo) / 8+r (hi)
    for (int r = 0; r < 8; ++r) {
        float* crow = C + (size_t)(mbase + r) * HIDDEN + nr;
        crow[0]  = acc0[r];
        crow[16] = acc1[r];
        crow[32] = acc2[r];
        crow[48] = acc3[r];
    }
}

// ---------------------------------------------------------------------------
// layer-1 edge aggregation: agg[dst] += dinv[s]*dinv[d] * h[src]   (64 feats)
// one wave per edge, 2 floats per lane
// ---------------------------------------------------------------------------
__global__ __launch_bounds__(256) void edge_agg1_kernel(const int* __restrict__ src,
                                                        const int* __restrict__ dst,
                                                        const float* __restrict__ dinv,
                                                        const float* __restrict__ h,
                                                        float* __restrict__ agg, int nE) {
    int e = blockIdx.x * 8 + (threadIdx.x >> 5);
    if (e >= nE) return;
    int lane = threadIdx.x & 31;
    int s = src[e], d = dst[e];
    float coeff = dinv[s] * dinv[d];
    const float* hs = h   + (size_t)s * HIDDEN + lane * 2;
    float*       ad = agg + (size_t)d * HIDDEN + lane * 2;
    atomic_add_f32(&ad[0], hs[0] * coeff);
    atomic_add_f32(&ad[1], hs[1] * coeff);
}

// ---------------------------------------------------------------------------
// h1_act = relu(agg + dinv[i]^2 * h1_pre + b1)   (self-loop folded in, in-place)
// ---------------------------------------------------------------------------
__global__ __launch_bounds__(256) void relu_kernel(const float* __restrict__ pre,
                                                   float* __restrict__ agg,
                                                   const float* __restrict__ dinv,
                                                   const float* __restrict__ b1) {
    int t = blockIdx.x * 256 + threadIdx.x;
    if (t >= N_NODES * HIDDEN) return;
    int node = t >> 6, f = t & 63;
    float di = dinv[node];
    float v = agg[t] + di * di * pre[t] + b1[f];
    agg[t] = v > 0.0f ? v : 0.0f;
}

// ---------------------------------------------------------------------------
// GEMM2: C[100000x16] = A[100000x64] * B[64x16]   (fp32 WMMA 16x16x4)
// ---------------------------------------------------------------------------
__global__ __launch_bounds__(256) void gemm2_kernel(const float* __restrict__ A,
                                                    const float* __restrict__ B,
                                                    float* __restrict__ C, int nTiles) {
    int wave = threadIdx.x >> 5;
    int lane = threadIdx.x & 31;
    int tile = blockIdx.x * 8 + wave;
    if (tile >= nTiles) return;
    int nr = lane & 15;
    int hi = lane >> 4;

    const float* arow = A + (size_t)(tile * 16 + nr) * HIDDEN + 2 * hi;
    v8f acc = {};
    for (int kk = 0; kk < HIDDEN; kk += 4) {
        v2f a = *(const v2f*)(arow + kk);
        const float* b0p = B + (size_t)(kk + 2 * hi) * N_CLASSES + nr;
        v2f b = { b0p[0], b0p[N_CLASSES] };
        acc = __builtin_amdgcn_wmma_f32_16x16x4_f32(false, a, false, b, (short)0, acc, false, false);
    }
    int mbase = tile * 16 + hi * 8;
    for (int r = 0; r < 8; ++r)
        C[(size_t)(mbase + r) * N_CLASSES + nr] = acc[r];
}

// ---------------------------------------------------------------------------
// layer-2 edge aggregation: out[dst] += dinv[s]*dinv[d] * h2[src]  (16 classes)
// one thread per (edge, class)
// ---------------------------------------------------------------------------
__global__ __launch_bounds__(256) void edge_agg2_kernel(const int* __restrict__ src,
                                                        const int* __restrict__ dst,
                                                        const float* __restrict__ dinv,
                                                        const float* __restrict__ h,
                                                        float* __restrict__ out, int nE) {
    int t = blockIdx.x * 256 + threadIdx.x;
    if (t >= nE * N_CLASSES) return;
    int e = t >> 4, c = t & 15;
    int s = src[e], d = dst[e];
    float coeff = dinv[s] * dinv[d];
    atomic_add_f32(&out[(size_t)d * N_CLASSES + c], coeff * h[(size_t)s * N_CLASSES + c]);
}

// ---------------------------------------------------------------------------
// out = log_softmax(out + dinv[i]^2 * h2_pre + b2)   (self-loop + bias folded)
// ---------------------------------------------------------------------------
__global__ __launch_bounds__(256) void logsoftmax_kernel(float* __restrict__ out,
                                                         const float* __restrict__ pre,
                                                         const float* __restrict__ dinv,
                                                         const float* __restrict__ b2) {
    int r = blockIdx.x * 256 + threadIdx.x;
    if (r >= N_NODES) return;
    float di = dinv[r];
    float di2 = di * di;
    float v[N_CLASSES];
    float m = -INFINITY;
    for (int c = 0; c < N_CLASSES; ++c) {
        float x = out[(size_t)r * N_CLASSES + c] + di2 * pre[(size_t)r * N_CLASSES + c] + b2[c];
        v[c] = x;
        m = fmaxf(m, x);
    }
    float s = 0.0f;
    for (int c = 0; c < N_CLASSES; ++c) s += expf(v[c] - m);
    float ls = logf(s);
    for (int c = 0; c < N_CLASSES; ++c)
        out[(size_t)r * N_CLASSES + c] = v[c] - m - ls;
}

// ---------------------------------------------------------------------------
extern "C" void kernel_launch(void* const* d_in, const int* in_sizes, int n_in,
                              void* d_out, int out_size, void* d_ws, size_t ws_size,
                              hipStream_t stream) {
    const float* x  = (const float*)d_in[0];
    const int*   ei = (const int*)d_in[1];
    const float* W1 = (const float*)d_in[2];
    const float* b1 = (const float*)d_in[3];
    const float* W2 = (const float*)d_in[4];
    const float* b2 = (const float*)d_in[5];
    float* out = (float*)d_out;

    int nE = in_sizes[1] / 2;
    const int* src = ei;
    const int* dst = ei + nE;

    // workspace layout
    float* dinv = (float*)d_ws;                              // 100000 (padded 102400)
    float* bufA = dinv + 102400;                             // h1_pre, later h2_pre
    float* bufB = bufA + (size_t)N_NODES * HIDDEN;           // h1_agg -> h1_act

    int nTiles = N_NODES / 16;                               // 6250, exact

    // zero accumulators + deg=1 (self-loops)
    init_kernel<<<(N_NODES * HIDDEN + 255) / 256, 256, 0, stream>>>(bufB, out, dinv);
    // degrees and symmetric norm
    degree_kernel<<<(nE + 255) / 256, 256, 0, stream>>>(dst, dinv, nE);
    rsqrt_kernel<<<(N_NODES + 255) / 256, 256, 0, stream>>>(dinv);
    // layer 1
    gemm1_kernel<<<(nTiles + 7) / 8, 256, 0, stream>>>(x, W1, bufA, nTiles);
    edge_agg1_kernel<<<(nE + 7) / 8, 256, 0, stream>>>(src, dst, dinv, bufA, bufB, nE);
    relu_kernel<<<(N_NODES * HIDDEN + 255) / 256, 256, 0, stream>>>(bufA, bufB, dinv, b1);
    // layer 2
    gemm2_kernel<<<(nTiles + 7) / 8, 256, 0, stream>>>(bufB, W2, bufA, nTiles);
    edge_agg2_kernel<<<(nE * N_CLASSES + 255) / 256, 256, 0, stream>>>(src, dst, dinv, bufA, out, nE);
    logsoftmax_kernel<<<(N_NODES + 255) / 256, 256, 0, stream>>>(out, bufA, dinv, b2);
}